// SPDDynamics_33732673142975
// MI455X (gfx1250) — compile-verified
//
#include <hip/hip_runtime.h>

typedef float v2f __attribute__((ext_vector_type(2)));
typedef float v8f __attribute__((ext_vector_type(8)));
typedef unsigned int u32x4 __attribute__((ext_vector_type(4)));
typedef int i32x4 __attribute__((ext_vector_type(4)));
typedef int i32x8 __attribute__((ext_vector_type(8)));

#define CDIM 64
#define HW   4096
#define HIDN 128
#define LDP  68                  // padded LDS row stride (floats): 16B aligned rows, bank-shift 4
#define MSZ  (CDIM * LDP)
#define TSTEP 0.05f
#define EPS0 1e-6f
#define NS_ITERS 11

#define EW_FOR(i) for (int i = tid; i < CDIM * CDIM; i += 256)

#if defined(__gfx1250__) && __has_builtin(__builtin_amdgcn_tensor_load_to_lds)
#define HAVE_TDM 1
#else
#define HAVE_TDM 0
#endif

#if HAVE_TDM
// Issue one TDM load of a 64-row x 64-float chunk (tensor row stride 4096 f32)
// into LDS at lds_byte_off, with hardware padding of 4 DWORDs every 64 DWORDs
// so rows land at stride LDP=68 floats. 2-D tensor -> groups 2/3 are NULL-like.
// This toolchain's builtin is the 6-arg form:
//   (u32x4 g0, i32x8 g1, i32x4 g2, i32x4 g3, i32x8 extra, i32 cpol)
__device__ __forceinline__ void tdm_issue_chunk(unsigned lds_byte_off, const float* gsrc)
{
    const unsigned long long ga = (unsigned long long)(uintptr_t)gsrc;
    u32x4 g0;
    g0.x = 1u;                                                   // count=1 (valid user D#)
    g0.y = lds_byte_off;                                         // lds_addr [63:32]
    g0.z = (unsigned)ga;                                         // global_addr[31:0]
    g0.w = (unsigned)((ga >> 32) & 0x01FFFFFFu) | (2u << 30);    // global_addr[56:32] | type=2
    i32x8 g1;
    g1[0] = (2 << 16)      // data_size = 4 bytes
          | (1 << 20)      // pad_enable
          | (5 << 22)      // pad_interval code 5 -> every 64 DWORDs
          | (3 << 25);     // pad_amount  code 3 -> 4 DWORDs pad (row stride 68 f32)
    g1[1] = (int)(4096u << 16);   // tensor_dim0[15:0] in [31:16]
    g1[2] = (int)(64u << 16);     // tensor_dim0 hi=0; tensor_dim1[15:0]=64 in [31:16]
    g1[3] = (int)(64u << 16);     // tensor_dim1 hi=0; tile_dim0=64 in [31:16]
    g1[4] = 64;                   // tile_dim1=64; tile_dim2=0
    g1[5] = 4096;                 // tensor_dim0_stride[31:0] = 4096 elements
    g1[6] = 0;                    // stride hi / tensor_dim1_stride lo
    g1[7] = 0;
    const i32x4 z4 = {0, 0, 0, 0};
    const i32x8 z8 = {0, 0, 0, 0, 0, 0, 0, 0};
    __builtin_amdgcn_tensor_load_to_lds(g0, g1, z4, z4, z8, 0);
}
#endif

// One 16x16x4 fp32 WMMA step with generic strides.
// A element (m,k) = A[m*sAm + k*sAk]; B element (k,n) = B[k*sBk + n*sBn].
__device__ __forceinline__ v8f tile_mma4(const float* A, int sAm, int sAk,
                                         const float* B, int sBk, int sBn,
                                         v8f acc, int lane)
{
    const int hf = lane >> 4;     // lane half selects K pair
    const int mn = lane & 15;     // row of A / col of B owned by this lane
    const int k0 = hf << 1;
#if __has_builtin(__builtin_amdgcn_wmma_f32_16x16x4_f32)
    v2f a, b;
    a.x = A[mn * sAm + (k0 + 0) * sAk];
    a.y = A[mn * sAm + (k0 + 1) * sAk];
    b.x = B[(k0 + 0) * sBk + mn * sBn];
    b.y = B[(k0 + 1) * sBk + mn * sBn];
    acc = __builtin_amdgcn_wmma_f32_16x16x4_f32(false, a, false, b,
                                                (short)0, acc, false, false);
#else
    #pragma unroll
    for (int i = 0; i < 8; ++i) {
        const int m = i + 8 * hf;
        float s = acc[i];
        for (int k = 0; k < 4; ++k)
            s += A[m * sAm + k * sAk] * B[k * sBk + mn * sBn];
        acc[i] = s;
    }
#endif
    return acc;
}

// C = A @ B for 64x64 matrices in LDS (row stride LDP). 8 waves x 2 tiles.
// Caller is responsible for barriers around this.
__device__ void mm64(const float* A, const float* B, float* C, int wave, int lane)
{
    const int hf = lane >> 4;
    const int mn = lane & 15;
    for (int t = wave; t < 16; t += 8) {
        const int tm = t >> 2, tn = t & 3;
        const float* At = A + tm * 16 * LDP;
        const float* Bt = B + tn * 16;
        v8f acc = {};
        #pragma unroll 4
        for (int k = 0; k < CDIM; k += 4)
            acc = tile_mma4(At + k, LDP, 1, Bt + k * LDP, LDP, 1, acc, lane);
        float* Cr = C + (tm * 16 + 8 * hf) * LDP + tn * 16 + mn;
        #pragma unroll
        for (int i = 0; i < 8; ++i) Cr[i * LDP] = acc[i];
    }
}

__global__ void __launch_bounds__(256)
spd_dyn_kernel(const float* __restrict__ x,
               const float* __restrict__ W1,
               const float* __restrict__ b1,
               const float* __restrict__ W2,
               const float* __restrict__ b2,
               const float* __restrict__ dscale_p,
               const float* __restrict__ noise,
               float* __restrict__ out)
{
    extern __shared__ float smem[];
    float* Mb[8];
    #pragma unroll
    for (int i = 0; i < 8; ++i) Mb[i] = smem + i * MSZ;
    float* sSum  = smem + 8 * MSZ;   // 64
    float* shid  = sSum + 64;        // 128
    float* sRed  = shid + 128;       // 256
    float* sScal = sRed + 256;       // 4

    const int tid  = threadIdx.x;
    const int wave = tid >> 5;
    const int lane = tid & 31;
    const int hf   = lane >> 4;
    const int mn   = lane & 15;
    const int b    = blockIdx.x;
    const float* xb = x + (size_t)b * (CDIM * HW);

    // ------- Stage 1+2 fused: single pass over x -------------------------
    // TDM double-buffered staging of 64x64 chunks; WMMA Gram accumulation
    // and per-channel sums both read the staged LDS tile (x read ONCE).
#if HAVE_TDM
    const unsigned bufoff0 = 5u * MSZ * 4u;      // dynamic-LDS base is 0
    if (wave == 0) tdm_issue_chunk(bufoff0, xb);
#endif
    v8f acc0 = {}, acc1 = {};
    float rs0 = 0.f, rs1 = 0.f, rs2 = 0.f, rs3 = 0.f;   // 4 independent row-sum accums
    const int t0 = wave * 2, t1 = wave * 2 + 1;
    const int srow = tid & 63, squart = tid >> 6;        // row-sum assignment
    for (int ci = 0; ci < 64; ++ci) {
        // pure arithmetic from the __shared__ base keeps addrspace(3)
        // provenance -> ds_load (not flat_load) in the WMMA feed loop
        float* sX = smem + (5 + (ci & 1)) * MSZ;
#if HAVE_TDM
        __syncthreads();   // everyone done computing on the other buffer
        if (wave == 0) {
            if (ci + 1 < 64) {
                tdm_issue_chunk(bufoff0 + (unsigned)(((ci + 1) & 1) * MSZ * 4),
                                xb + (ci + 1) * 64);
                __builtin_amdgcn_s_wait_tensorcnt((short)1);  // chunk ci complete
            } else {
                __builtin_amdgcn_s_wait_tensorcnt((short)0);
            }
        }
        __syncthreads();   // chunk ci visible to all waves
#else
        __syncthreads();
        {   // cooperative fallback: 64 rows x 64 k, float4 per thread x4
            const int row = tid >> 2, q = tid & 3;
            const float* src = xb + row * HW + ci * 64 + q * 16;
            float* dst = sX + row * LDP + q * 16;
            *(float4*)(dst + 0)  = *(const float4*)(src + 0);
            *(float4*)(dst + 4)  = *(const float4*)(src + 4);
            *(float4*)(dst + 8)  = *(const float4*)(src + 8);
            *(float4*)(dst + 12) = *(const float4*)(src + 12);
        }
        __syncthreads();
#endif
        {   // per-channel partial sums: 4 independent chains, co-issue w/ WMMA
            const float* rr = sX + srow * LDP + squart * 16;
            #pragma unroll
            for (int u = 0; u < 4; ++u) {
                rs0 += rr[u];
                rs1 += rr[u + 4];
                rs2 += rr[u + 8];
                rs3 += rr[u + 12];
            }
        }
        {   // Gram tiles via WMMA (each wave owns 2 of 16 output tiles)
            const float* A0 = sX + (t0 >> 2) * 16 * LDP;
            const float* B0 = sX + (t0 & 3) * 16 * LDP;   // transposed access
            const float* A1 = sX + (t1 >> 2) * 16 * LDP;
            const float* B1 = sX + (t1 & 3) * 16 * LDP;
            #pragma unroll 4
            for (int kk = 0; kk < 64; kk += 4) {
                acc0 = tile_mma4(A0 + kk, LDP, 1, B0 + kk, 1, LDP, acc0, lane);
                acc1 = tile_mma4(A1 + kk, LDP, 1, B1 + kk, 1, LDP, acc1, lane);
            }
        }
    }
    sRed[tid] = (rs0 + rs1) + (rs2 + rs3);
    __syncthreads();
    if (tid < 64)
        sSum[tid] = sRed[tid] + sRed[tid + 64] + sRed[tid + 128] + sRed[tid + 192];
    __syncthreads();
    {   // cov = (G - sum sum^T / HW) / (HW-1) + eps*I   -> Mb[0]
        const float inv_hw = 1.f / (float)HW;
        const float inv_n1 = 1.f / (float)(HW - 1);
        #pragma unroll
        for (int u = 0; u < 2; ++u) {
            const int t  = u ? t1 : t0;
            const v8f av = u ? acc1 : acc0;
            const int tm = t >> 2, tn = t & 3;
            #pragma unroll
            for (int i = 0; i < 8; ++i) {
                const int m = tm * 16 + i + 8 * hf;
                const int n = tn * 16 + mn;
                float cv = (av[i] - sSum[m] * sSum[n] * inv_hw) * inv_n1;
                if (m == n) cv += EPS0;
                Mb[0][m * LDP + n] = cv;
            }
        }
    }
    __syncthreads();

    // ---------------- Stage 3: drift MLP, symmetrized, scaled by T -------
    if (tid < HIDN) {
        float a = b1[tid];
        for (int i = 0; i < CDIM * CDIM; ++i)
            a += Mb[0][(i >> 6) * LDP + (i & 63)] * W1[i * HIDN + tid];
        shid[tid] = a > 0.f ? a : 0.f;
    }
    __syncthreads();
    {
        float dr[16];
        const int base = tid * 16;
        #pragma unroll
        for (int u = 0; u < 16; ++u) dr[u] = b2[base + u];
        for (int j = 0; j < HIDN; ++j) {
            const float hv = shid[j];
            const float* w2r = W2 + j * (CDIM * CDIM) + base;
            #pragma unroll
            for (int u = 0; u < 16; ++u) dr[u] += hv * w2r[u];
        }
        #pragma unroll
        for (int u = 0; u < 16; ++u) {
            const int i = base + u;
            Mb[5][(i >> 6) * LDP + (i & 63)] = dr[u];
        }
    }
    __syncthreads();
    EW_FOR(i) {
        const int r = i >> 6, c = i & 63;
        Mb[7][r * LDP + c] = (TSTEP * 0.5f) * (Mb[5][r * LDP + c] + Mb[5][c * LDP + r]);
    }
    __syncthreads();

    // ---------------- Stage 4: Newton-Schulz: Y->(P/tr)^1/2, Z->(P/tr)^-1/2
    if (tid == 0) {
        float tr = 0.f;
        for (int i = 0; i < CDIM; ++i) tr += Mb[0][i * LDP + i];
        sScal[0] = fmaxf(tr, 1e-20f);
    }
    __syncthreads();
    const float trc = sScal[0];
    EW_FOR(i) {
        const int r = i >> 6, c = i & 63;
        Mb[1][r * LDP + c] = Mb[0][r * LDP + c] / trc;
        Mb[2][r * LDP + c] = (r == c) ? 1.f : 0.f;
    }
    __syncthreads();
    float *Y = Mb[1], *Z = Mb[2], *Yn = Mb[3], *Zn = Mb[4], *Tt = Mb[5], *Wk = Mb[6];
    for (int it = 0; it < NS_ITERS; ++it) {
        mm64(Z, Y, Wk, wave, lane);
        __syncthreads();
        EW_FOR(i) {
            const int r = i >> 6, c = i & 63;
            float v = -0.5f * Wk[r * LDP + c];
            if (r == c) v += 1.5f;
            Tt[r * LDP + c] = v;
        }
        __syncthreads();
        mm64(Y, Tt, Yn, wave, lane);
        mm64(Tt, Z, Zn, wave, lane);
        __syncthreads();
        float* tp = Y; Y = Yn; Yn = tp;
        tp = Z; Z = Zn; Zn = tp;
    }

    // ---------------- Stage 5: middle = (Z D Z)/tr, pre-scaled by 1/16 ---
    float* Tmp1 = Yn;   // free ping-pong buffers
    float* Mid  = Zn;
    mm64(Z, Mb[7], Tmp1, wave, lane);
    __syncthreads();
    mm64(Tmp1, Z, Mid, wave, lane);
    __syncthreads();
    const float msc = 1.f / (trc * 16.f);
    EW_FOR(i) { const int r = i >> 6, c = i & 63; Mid[r * LDP + c] *= msc; }
    __syncthreads();

    // ---------------- Stage 6: expm via Taylor-Horner + 4 squarings ------
    float* Aexp = Tt;
    float* Texp = Wk;
    EW_FOR(i) {
        const int r = i >> 6, c = i & 63;
        Aexp[r * LDP + c] = Mid[r * LDP + c] * (1.f / 6.f) + (r == c ? 1.f : 0.f);
    }
    __syncthreads();
    for (int k = 5; k >= 1; --k) {
        mm64(Mid, Aexp, Texp, wave, lane);
        __syncthreads();
        const float ik = 1.f / (float)k;
        EW_FOR(i) {
            const int r = i >> 6, c = i & 63;
            Aexp[r * LDP + c] = Texp[r * LDP + c] * ik + (r == c ? 1.f : 0.f);
        }
        __syncthreads();
    }
    for (int s = 0; s < 4; ++s) {
        mm64(Aexp, Aexp, Texp, wave, lane);
        __syncthreads();
        float* tp = Aexp; Aexp = Texp; Texp = tp;
    }

    // ---------------- Stage 7: C_new = tr*(Y E Y) + noise; Bm = C_new-eps*I
    mm64(Y, Aexp, Tmp1, wave, lane);
    __syncthreads();
    float* Bm = Mid;
    mm64(Tmp1, Y, Bm, wave, lane);
    __syncthreads();
    const float nsc = dscale_p[0] * sqrtf(TSTEP) * 0.5f;
    const float* nb = noise + (size_t)b * (CDIM * CDIM);
    EW_FOR(i) {
        const int r = i >> 6, c = i & 63;
        float v = Bm[r * LDP + c] * trc + nsc * (nb[r * 64 + c] + nb[c * 64 + r]);
        if (r == c) v -= EPS0;
        Bm[r * LDP + c] = v;
    }
    __syncthreads();

    // ---------------- Stage 8: SPD projection: eps*I + (Bm + |Bm|)/2 -----
    float* A2 = Tmp1;
    mm64(Bm, Bm, A2, wave, lane);
    __syncthreads();
    if (tid == 0) {
        float tr = 0.f;
        for (int i = 0; i < CDIM; ++i) tr += A2[i * LDP + i];
        sScal[1] = fmaxf(tr, 1e-20f);
    }
    __syncthreads();
    const float tr2 = sScal[1];
    float *Y2 = Mb[3], *Z2 = Mb[4], *Y2n = Mb[5], *Z2n = Mb[6], *Tt2 = Mb[7], *W2b = Mb[0];
    EW_FOR(i) {
        const int r = i >> 6, c = i & 63;
        Y2[r * LDP + c] = A2[r * LDP + c] / tr2;
        Z2[r * LDP + c] = (r == c) ? 1.f : 0.f;
    }
    __syncthreads();
    for (int it = 0; it < NS_ITERS; ++it) {
        mm64(Z2, Y2, W2b, wave, lane);
        __syncthreads();
        EW_FOR(i) {
            const int r = i >> 6, c = i & 63;
            float v = -0.5f * W2b[r * LDP + c];
            if (r == c) v += 1.5f;
            Tt2[r * LDP + c] = v;
        }
        __syncthreads();
        mm64(Y2, Tt2, Y2n, wave, lane);
        mm64(Tt2, Z2, Z2n, wave, lane);
        __syncthreads();
        float* tp = Y2; Y2 = Y2n; Y2n = tp;
        tp = Z2; Z2 = Z2n; Z2n = tp;
    }
    const float s2 = sqrtf(tr2);
    float* ob = out + (size_t)b * (CDIM * CDIM);
    EW_FOR(i) {
        const int r = i >> 6, c = i & 63;
        float v = 0.5f * (Bm[r * LDP + c] + s2 * Y2[r * LDP + c]);
        if (r == c) v += EPS0;
        ob[r * 64 + c] = v;
    }
}

extern "C" void kernel_launch(void* const* d_in, const int* in_sizes, int n_in,
                              void* d_out, int out_size, void* d_ws, size_t ws_size,
                              hipStream_t stream)
{
    const float* x  = (const float*)d_in[0];
    const float* W1 = (const float*)d_in[1];
    const float* b1 = (const float*)d_in[2];
    const float* W2 = (const float*)d_in[3];
    const float* b2 = (const float*)d_in[4];
    const float* ds = (const float*)d_in[5];
    const float* nz = (const float*)d_in[6];
    float* out = (float*)d_out;

    const int B = in_sizes[0] / (CDIM * HW);   // 256 batch SPD matrices
    const size_t shmem = (size_t)(8 * MSZ + 64 + 128 + 256 + 4) * sizeof(float);

    spd_dyn_kernel<<<dim3(B), dim3(256), shmem, stream>>>(x, W1, b1, W2, b2, ds, nz, out);
}